// GenericClosing2D_49314814492673
// MI455X (gfx1250) — compile-verified
//
#include <hip/hip_runtime.h>
#include <cstdint>

namespace {
constexpr int Bn = 8, Cn = 32, Hn = 512, Wn = 512;
constexpr int TILE  = 32;
constexpr int ITILE = TILE + 12;   // 44: input tile with closing halo (6 each side)
constexpr int DTILE = TILE + 6;    // 38: dilated tile with erosion halo (3 each side)
constexpr int ISTR  = 44;          // natural TDM row stride; 176B rows stay 16B aligned
constexpr int DSTR  = 40;
constexpr int NT    = 256;
}

typedef unsigned int u32;
typedef __attribute__((ext_vector_type(4))) u32 u32x4;
typedef __attribute__((ext_vector_type(8))) u32 u32x8;

__global__ __launch_bounds__(NT)
void closing7x7_kernel(const float* __restrict__ x,
                       const float* __restrict__ kern,
                       float* __restrict__ out)
{
    __shared__ float s_in[ITILE * ISTR + 8];  // +8 pad: OOB-but-discarded tail reads
    __shared__ float s_dil[DTILE * DSTR];

    const int tid   = threadIdx.x;
    const int plane = blockIdx.z;             // b*C + c
    const int c     = plane & (Cn - 1);
    const int tx0   = blockIdx.x * TILE;
    const int ty0   = blockIdx.y * TILE;
    const int gx0   = tx0 - 6;
    const int gy0   = ty0 - 6;

    const float* __restrict__ xp = x   + (size_t)plane * (Hn * Wn);
    float* __restrict__       op = out + (size_t)plane * (Hn * Wn);

    const float NEG = -__builtin_inff();
    const float POS =  __builtin_inff();

    // 49 kernel taps: uniform address -> scalar loads / SGPRs
    float kv[49];
    {
        const float* kc = kern + c * 49;
#pragma unroll
        for (int t = 0; t < 49; ++t) kv[t] = kc[t];
    }

    // ---- Stage 1: stage 44x44 input tile into LDS ----
    const bool interior = (gx0 >= 0) & (gy0 >= 0) &
                          ((gx0 + ITILE) <= Wn) & ((gy0 + ITILE) <= Hn);
    if (interior) {
        // Tensor Data Mover: one DMA descriptor moves the whole 2D tile.
        if (tid < 32) {   // wave 0 only (EXEC ignored by tensor ops)
            uint64_t ga = (uint64_t)(uintptr_t)(xp + (size_t)gy0 * Wn + gx0);
            u32 lds = (u32)(uintptr_t)&s_in[0];     // low 32b of flat LDS ptr = LDS offset
            u32x4 g0;
            g0[0] = 0x1u;                                   // count=1, load, no gather
            g0[1] = lds;                                    // lds_addr
            g0[2] = (u32)ga;                                // global_addr[31:0]
            g0[3] = (u32)((ga >> 32) & 0x01FFFFFFu) | 0x80000000u; // addr[56:32] | type=2
            u32x8 g1;
            g1[0] = 0x2u << 16;                             // data_size=4B; no pad/iter/mcast
            g1[1] = ((u32)Wn & 0xFFFFu) << 16;              // tensor_dim0[15:0]=512
            g1[2] = ((u32)Hn & 0xFFFFu) << 16;              // tensor_dim1[15:0]=512
            g1[3] = (u32)ITILE << 16;                       // tile_dim0=44
            g1[4] = (u32)ITILE;                             // tile_dim1=44, tile_dim2=0
            g1[5] = (u32)Wn;                                // tensor_dim0_stride=512
            g1[6] = 0u;
            g1[7] = 0u;
            asm volatile("tensor_load_to_lds %0, %1"
                         :: "s"(g0), "s"(g1) : "memory");
            __builtin_amdgcn_s_wait_tensorcnt(0);
        }
        __syncthreads();
    } else {
        // Edge tiles: per-lane async loads with clamped addresses, then -inf patch.
#pragma unroll
        for (int it = 0; it < (ITILE * ITILE + NT - 1) / NT; ++it) {
            int idx = tid + it * NT;
            if (idx < ITILE * ITILE) {
                int rr = idx / ITILE;
                int cc = idx - rr * ITILE;
                int gy = gy0 + rr, gx = gx0 + cc;
                int cy = gy < 0 ? 0 : (gy >= Hn ? Hn - 1 : gy);
                int cx = gx < 0 ? 0 : (gx >= Wn ? Wn - 1 : gx);
                const float* gp = xp + cy * Wn + cx;
                unsigned loff = (unsigned)(uintptr_t)&s_in[rr * ISTR + cc];
                asm volatile("global_load_async_to_lds_b32 %0, %1, off"
                             :: "v"(loff), "v"(gp) : "memory");
            }
        }
        asm volatile("s_wait_asynccnt 0x0" ::: "memory");
        __syncthreads();
        for (int idx = tid; idx < ITILE * ITILE; idx += NT) {
            int rr = idx / ITILE;
            int cc = idx - rr * ITILE;
            int gy = gy0 + rr, gx = gx0 + cc;
            if ((unsigned)gy >= (unsigned)Hn || (unsigned)gx >= (unsigned)Wn)
                s_in[rr * ISTR + cc] = NEG;
        }
        __syncthreads();
    }

    // ---- Stage 2: max-plus dilation -> s_dil (38x38); +inf outside image ----
    // dil[r][cc] corresponds to image (ty0+r-3, tx0+cc-3); taps use flipped kernel.
#pragma unroll
    for (int it = 0; it < 2; ++it) {
        int u = tid + it * NT;          // 380 units of 4 consecutive columns
        if (u < 380) {
            int r  = u / 10;
            int cg = u - r * 10;
            int c0 = cg * 4;
            float m0 = NEG, m1 = NEG, m2 = NEG, m3 = NEG;
#pragma unroll
            for (int i = 0; i < 7; ++i) {
                float rv[10];
#pragma unroll
                for (int j = 0; j < 10; ++j) rv[j] = s_in[(r + i) * ISTR + c0 + j];
#pragma unroll
                for (int j = 0; j < 7; ++j) {
                    float kd = kv[48 - (i * 7 + j)];   // kernel[6-i][6-j]
                    m0 = fmaxf(m0, rv[j + 0] + kd);
                    m1 = fmaxf(m1, rv[j + 1] + kd);
                    m2 = fmaxf(m2, rv[j + 2] + kd);
                    m3 = fmaxf(m3, rv[j + 3] + kd);
                }
            }
            int  dy     = ty0 + r - 3;
            bool rowOOB = (unsigned)dy >= (unsigned)Hn;
            float mm[4] = {m0, m1, m2, m3};
#pragma unroll
            for (int q = 0; q < 4; ++q) {
                int cc = c0 + q;
                if (cc < DTILE) {
                    int dx = tx0 + cc - 3;
                    float v = (rowOOB || (unsigned)dx >= (unsigned)Wn) ? POS : mm[q];
                    s_dil[r * DSTR + cc] = v;
                }
            }
        }
    }
    __syncthreads();

    // ---- Stage 3: min-minus erosion -> global, 4-wide vectorized store ----
    {
        int ty  = tid >> 3;        // 0..31
        int txb = (tid & 7) * 4;   // 0,4,...,28
        float m0 = POS, m1 = POS, m2 = POS, m3 = POS;
#pragma unroll
        for (int i = 0; i < 7; ++i) {
            float rv[10];
#pragma unroll
            for (int j = 0; j < 10; ++j) rv[j] = s_dil[(ty + i) * DSTR + txb + j];
#pragma unroll
            for (int j = 0; j < 7; ++j) {
                float ke = kv[i * 7 + j];
                m0 = fminf(m0, rv[j + 0] - ke);
                m1 = fminf(m1, rv[j + 1] - ke);
                m2 = fminf(m2, rv[j + 2] - ke);
                m3 = fminf(m3, rv[j + 3] - ke);
            }
        }
        float4 o = make_float4(m0, m1, m2, m3);
        *reinterpret_cast<float4*>(op + (size_t)(ty0 + ty) * Wn + tx0 + txb) = o;
    }
}

extern "C" void kernel_launch(void* const* d_in, const int* in_sizes, int n_in,
                              void* d_out, int out_size, void* d_ws, size_t ws_size,
                              hipStream_t stream) {
    const float* x = (const float*)d_in[0];   // (B, C, H, W) f32
    const float* k = (const float*)d_in[1];   // (C, 7, 7)    f32
    float* o = (float*)d_out;                 // (B, C, H, W) f32
    (void)in_sizes; (void)n_in; (void)out_size; (void)d_ws; (void)ws_size;
    dim3 grid(Wn / TILE, Hn / TILE, Bn * Cn); // (16, 16, 256)
    dim3 block(NT);
    hipLaunchKernelGGL(closing7x7_kernel, grid, block, 0, stream, x, k, o);
}